// PGNNLayer_17746804867497
// MI455X (gfx1250) — compile-verified
//
#include <hip/hip_runtime.h>

typedef __attribute__((ext_vector_type(2))) float v2f;
typedef __attribute__((ext_vector_type(8))) float v8f;

#define NROW 32768
#define KNBR 32
#define DFEAT 128

// ---------------------------------------------------------------------------
// Kernel 1: y[n] = dot(x[n,:], w[:])  for both x1 and x2 (blockIdx.y selects).
// Uses V_WMMA_F32_16X16X4_F32: 16 rows of x as the A-matrix M-dim, D swept in
// K=4 chunks (32 WMMAs), w broadcast into every column of B, so each column of
// the 16x16 f32 accumulator holds the same 16 dots. Extract from lanes 0 / 16
// per the documented C/D layout (VGPR v: M=v for lanes 0-15, M=v+8 for 16-31).
// A-matrix 16x4 f32 layout (ISA 7.12.2): lanes 0-15 hold K=0(,1), lanes 16-31
// hold K=2(,3); VGPR0=K even slot, VGPR1=K odd slot.
// ---------------------------------------------------------------------------
__global__ __launch_bounds__(256) void project_wmma(
    const float* __restrict__ x1, const float* __restrict__ x2,
    const float* __restrict__ w,
    float* __restrict__ y1, float* __restrict__ y2)
{
    const float* x = (blockIdx.y == 0) ? x1 : x2;
    float*       y = (blockIdx.y == 0) ? y1 : y2;

    const int wave = threadIdx.x >> 5;     // 8 waves / block
    const int lane = threadIdx.x & 31;
    const int half = lane >> 4;            // 0 -> K={0,1}, 1 -> K={2,3}
    const int m    = lane & 15;            // row within 16-row tile
    const int row0 = (blockIdx.x * 8 + wave) * 16;

    const float* xr = x + (size_t)(row0 + m) * DFEAT + half * 2;
    const float* wr = w + half * 2;

    v8f c = {0.f, 0.f, 0.f, 0.f, 0.f, 0.f, 0.f, 0.f};

    #pragma unroll 4
    for (int d0 = 0; d0 < DFEAT; d0 += 4) {
        v2f a, b;
        a.x = xr[d0];
        a.y = xr[d0 + 1];
        b.x = wr[d0];          // same w chunk in all 16 B columns
        b.y = wr[d0 + 1];
        c = __builtin_amdgcn_wmma_f32_16x16x4_f32(
                /*neg_a=*/false, a, /*neg_b=*/false, b,
                /*c_mod=*/(short)0, c, /*reuse_a=*/false, /*reuse_b=*/false);
    }

    // Every column identical; lane 0 writes M=0..7, lane 16 writes M=8..15.
    if ((lane & 15) == 0) {
        float* yo = y + row0 + half * 8;
        #pragma unroll
        for (int v = 0; v < 8; ++v) yo[v] = c[v];
    }
}

// ---------------------------------------------------------------------------
// Kernel 2: one n per wave32.
//   out_pos[n,k]    = dmax[n,k] * y[idx[n,k]] + b        (lane k)
//   out_struct[n,:] = (1/K) * sum_k dmax[n,k] * x[idx[n,k],:]
// Each k-iteration the wave loads one gathered 512B row (lane -> float4,
// global_load_b128, fully coalesced; x is L2-resident) and FMA-accumulates.
// idx/dmax are wave-uniform per k -> broadcast via __shfl (ds_bpermute).
// ---------------------------------------------------------------------------
__global__ __launch_bounds__(256) void gather_fuse(
    const float* __restrict__ x,   const int* __restrict__ idx,
    const float* __restrict__ dmax,const float* __restrict__ y,
    const float* __restrict__ b_out,
    float* __restrict__ out_pos,   float* __restrict__ out_struct)
{
    const int wave = threadIdx.x >> 5;
    const int lane = threadIdx.x & 31;
    const int n    = blockIdx.x * 8 + wave;

    const float bb = b_out[0];

    const size_t eb = (size_t)n * KNBR + lane;
    const int   ik = idx[eb];
    const float dk = dmax[eb];

    // position output: scalar gather of precomputed projection
    out_pos[eb] = dk * y[ik] + bb;

    // structural output: weighted mean of gathered rows
    const int dcol = lane * 4;
    float ax = 0.f, ay = 0.f, az = 0.f, aw = 0.f;

    #pragma unroll 8
    for (int k = 0; k < KNBR; ++k) {
        const int   ikk = __shfl(ik, k, 32);
        const float dkk = __shfl(dk, k, 32);
        const float4 r = *(const float4*)(x + (size_t)ikk * DFEAT + dcol);
        ax += dkk * r.x;
        ay += dkk * r.y;
        az += dkk * r.z;
        aw += dkk * r.w;
    }

    const float s = 1.0f / (float)KNBR;
    float4* os = (float4*)(out_struct + (size_t)n * DFEAT + dcol);
    *os = make_float4(ax * s, ay * s, az * s, aw * s);
}

// ---------------------------------------------------------------------------
extern "C" void kernel_launch(void* const* d_in, const int* in_sizes, int n_in,
                              void* d_out, int out_size, void* d_ws, size_t ws_size,
                              hipStream_t stream)
{
    const float* x1  = (const float*)d_in[0];
    const int*   id1 = (const int*)  d_in[1];
    const float* dm1 = (const float*)d_in[2];
    const float* x2  = (const float*)d_in[3];
    const int*   id2 = (const int*)  d_in[4];
    const float* dm2 = (const float*)d_in[5];
    const float* w   = (const float*)d_in[6];
    const float* b   = (const float*)d_in[7];

    float* out = (float*)d_out;
    float* op1 = out;                                   // [N,K]
    float* os1 = op1 + (size_t)NROW * KNBR;             // [N,D]
    float* op2 = os1 + (size_t)NROW * DFEAT;            // [N,K]
    float* os2 = op2 + (size_t)NROW * KNBR;             // [N,D]

    float* y1 = (float*)d_ws;                           // [N]
    float* y2 = y1 + NROW;                              // [N]

    // Projection: 2048 row-tiles per input, 128 rows (8 waves) per block.
    dim3 gproj(NROW / 128, 2);
    project_wmma<<<gproj, 256, 0, stream>>>(x1, x2, w, y1, y2);

    // Gather-reduce: one n per wave, 8 waves per block.
    dim3 gg(NROW / 8);
    gather_fuse<<<gg, 256, 0, stream>>>(x1, id1, dm1, y1, b, op1, os1);
    gather_fuse<<<gg, 256, 0, stream>>>(x2, id2, dm2, y2, b, op2, os2);
}